// PoolGraph_65343632441865
// MI455X (gfx1250) — compile-verified
//
#include <hip/hip_runtime.h>
#include <hip/hip_bf16.h>
#include <stdint.h>

#ifndef __has_builtin
#define __has_builtin(x) 0
#endif

typedef float v2f __attribute__((ext_vector_type(2)));
typedef float v8f __attribute__((ext_vector_type(8)));
typedef unsigned int tdm_v4u __attribute__((ext_vector_type(4)));
typedef int tdm_v8i __attribute__((ext_vector_type(8)));
typedef int tdm_v4i __attribute__((ext_vector_type(4)));

#define NB     4
#define NPERG  1024
#define NTOT   4096
#define KKEEP  512
#define NKEPT  2048
#define DF     128
#define NE     65536      // total directed edges
#define NEV    16         // subspace width (WMMA N)
#define KP     1024       // k-panel rows staged in LDS (KP*NEV*4 = 64KB)
#define NITER  48
#define ALPHA_ 0.5f

#if defined(__HIP_DEVICE_COMPILE__) && \
    __has_builtin(__builtin_amdgcn_tensor_load_to_lds) && \
    __has_builtin(__builtin_amdgcn_s_wait_tensorcnt)
#define HAVE_TDM 1
#endif

#ifdef HAVE_TDM
// 1-D contiguous TDM load: ndwords dwords from gsrc -> LDS at lds_dst.
__device__ __forceinline__ void tdm_load_1d(void* lds_dst, const void* gsrc, unsigned ndwords) {
  unsigned long long ga = (unsigned long long)(uintptr_t)gsrc;
  tdm_v4u g0;
  g0.x = 1u;                                                   // count=1 (valid user D#)
  g0.y = (unsigned)(uintptr_t)lds_dst;                         // lds_addr (LDS byte offset)
  g0.z = (unsigned)(ga & 0xFFFFFFFFull);                       // global_addr[31:0]
  g0.w = (unsigned)((ga >> 32) & 0x01FFFFFFull) | (2u << 30);  // global_addr[56:32] | type=2
  tdm_v8i g1;
  g1[0] = (int)(2u << 16);                                     // data_size=2 (4 bytes)
  g1[1] = (int)((ndwords & 0xFFFFu) << 16);                    // tensor_dim0[15:0]
  g1[2] = (int)((ndwords >> 16) | (1u << 16));                 // tensor_dim0[31:16] | tensor_dim1=1
  g1[3] = (int)((ndwords & 0xFFFFu) << 16);                    // tile_dim0
  g1[4] = 1;                                                   // tile_dim1=1
  g1[5] = (int)ndwords;                                        // tensor_dim0_stride[31:0]
  g1[6] = 0; g1[7] = 0;
  tdm_v4i gz = {0, 0, 0, 0};
#if __clang_major__ >= 23
  tdm_v8i gz8 = {0, 0, 0, 0, 0, 0, 0, 0};
  __builtin_amdgcn_tensor_load_to_lds(g0, g1, gz, gz, gz8, 0);
#else
  __builtin_amdgcn_tensor_load_to_lds(g0, g1, gz, gz, 0);
#endif
  __builtin_amdgcn_s_wait_tensorcnt(0);
}
#endif

// ---------------- GCN scoring ----------------

__global__ void k_deg_gcn(const int* __restrict__ col, float* __restrict__ deg) {
  int e = blockIdx.x * blockDim.x + threadIdx.x;
  if (e < NE) atomicAdd(&deg[col[e]], 1.0f);
}

__global__ void k_dinv_gcn(const float* __restrict__ deg, float* __restrict__ dinv) {
  int i = blockIdx.x * blockDim.x + threadIdx.x;
  if (i < NTOT) dinv[i] = rsqrtf(deg[i] + 1.0f);   // +1 self loop
}

__global__ void k_h(const float* __restrict__ x, const float* __restrict__ W,
                    float* __restrict__ h) {
  int wave = (blockIdx.x * blockDim.x + threadIdx.x) >> 5;
  int lane = threadIdx.x & 31;
  if (wave >= NTOT) return;
  const float* xr = x + (size_t)wave * DF;
  float s = 0.f;
  for (int d = lane; d < DF; d += 32) s += xr[d] * W[d];
  for (int off = 16; off; off >>= 1) s += __shfl_down(s, off, 32);
  if (lane == 0) h[wave] = s;
}

__global__ void k_gcn_scatter(const int* __restrict__ row, const int* __restrict__ col,
                              const float* __restrict__ dinv, const float* __restrict__ h,
                              float* __restrict__ acc) {
  int e = blockIdx.x * blockDim.x + threadIdx.x;
  if (e < NE) atomicAdd(&acc[col[e]], dinv[row[e]] * dinv[col[e]] * h[row[e]]);
}

__global__ void k_score(const float* __restrict__ acc, const float* __restrict__ dinv,
                        const float* __restrict__ h, const float* __restrict__ b,
                        float* __restrict__ score) {
  int i = blockIdx.x * blockDim.x + threadIdx.x;
  if (i < NTOT) score[i] = tanhf(acc[i] + dinv[i] * dinv[i] * h[i] + b[0]);
}

// ---------------- per-graph top-k (bitonic, 1024 elems, 512 threads) ----------------

__global__ void k_topk(const float* __restrict__ score, int* __restrict__ perm) {
  __shared__ float ss[NPERG];
  __shared__ int   si[NPERG];
  int g = blockIdx.x, t = threadIdx.x;
  for (int i = t; i < NPERG; i += 512) { ss[i] = score[g * NPERG + i]; si[i] = i; }
  __syncthreads();
  for (unsigned k = 2; k <= NPERG; k <<= 1) {
    for (unsigned j = k >> 1; j > 0; j >>= 1) {
      for (unsigned i = t; i < NPERG; i += 512) {
        unsigned ixj = i ^ j;
        if (ixj > i) {
          float s1 = ss[i], s2 = ss[ixj];
          int   i1 = si[i], i2 = si[ixj];
          bool before2_1 = (s2 > s1) || (s2 == s1 && i2 < i1); // elem[ixj] ranks earlier
          bool dir = ((i & k) == 0);
          if (before2_1 == dir) { ss[i] = s2; ss[ixj] = s1; si[i] = i2; si[ixj] = i1; }
        }
      }
      __syncthreads();
    }
  }
  for (int i = t; i < KKEEP; i += 512) perm[g * KKEEP + i] = g * NPERG + si[i];
}

__global__ void k_gather(const float* __restrict__ x, const float* __restrict__ score,
                         const int* __restrict__ perm, float* __restrict__ xout) {
  int j = blockIdx.x, t = threadIdx.x;
  int p = perm[j];
  xout[(size_t)j * DF + t] = x[(size_t)p * DF + t] * score[p];
}

__global__ void k_newid_init(int* __restrict__ nid) {
  int i = blockIdx.x * blockDim.x + threadIdx.x;
  if (i < NTOT) nid[i] = -1;
}
__global__ void k_newid_set(const int* __restrict__ perm, int* __restrict__ nid) {
  int j = blockIdx.x * blockDim.x + threadIdx.x;
  if (j < NKEPT) nid[perm[j]] = j;
}

// ---------------- deterministic edge compaction (flags -> scan -> scatter) ----------------

__global__ void k_flags(const int* __restrict__ row, const int* __restrict__ col,
                        const int* __restrict__ nid, int* __restrict__ flags) {
  int e = blockIdx.x * blockDim.x + threadIdx.x;
  if (e < NE) flags[e] = (nid[row[e]] >= 0 && nid[col[e]] >= 0) ? 1 : 0;
}

__global__ void k_scan(const int* __restrict__ flags, int* __restrict__ offs) {
  __shared__ int sh[1024];
  int t = threadIdx.x;
  int run = 0;
  for (int base = 0; base < NE; base += 1024) {
    int v = flags[base + t];
    sh[t] = v; __syncthreads();
    for (int off = 1; off < 1024; off <<= 1) {
      int add = (t >= off) ? sh[t - off] : 0;
      __syncthreads();
      sh[t] += add;
      __syncthreads();
    }
    offs[base + t] = run + sh[t] - v;
    int tot = sh[1023];
    __syncthreads();
    run += tot;
  }
  if (t == 0) offs[NE] = run;
}

__global__ void k_scatter_edges(const int* __restrict__ row, const int* __restrict__ col,
                                const int* __restrict__ nid, const int* __restrict__ flags,
                                const int* __restrict__ offs, int* __restrict__ er,
                                int* __restrict__ ec, float* __restrict__ out_edges) {
  int e = blockIdx.x * blockDim.x + threadIdx.x;
  if (e >= NE || !flags[e]) return;
  int pos = offs[e], Ecnt = offs[NE];
  int r = nid[row[e]], c = nid[col[e]];
  er[pos] = r; ec[pos] = c;
  out_edges[pos] = (float)r;
  out_edges[(size_t)Ecnt + pos] = (float)c;
}

// ---------------- Fiedler: build L, subspace iteration with WMMA ----------------

__global__ void k_deg_edges(const int* __restrict__ rows, const int* __restrict__ pcnt,
                            int fixed_cnt, float* __restrict__ deg) {
  int e = blockIdx.x * blockDim.x + threadIdx.x;
  int cnt = pcnt ? pcnt[0] : fixed_cnt;
  if (e < cnt) atomicAdd(&deg[rows[e]], 1.0f);
}

__global__ void k_dinv0(const float* __restrict__ deg, float* __restrict__ dinv, int n) {
  int i = blockIdx.x * blockDim.x + threadIdx.x;
  if (i < n) dinv[i] = deg[i] > 0.f ? rsqrtf(deg[i]) : 0.f;
}

__global__ void k_L_edges(const int* __restrict__ rows, const int* __restrict__ cols,
                          const int* __restrict__ pcnt, int fixed_cnt,
                          const float* __restrict__ dinv, float* __restrict__ L, int n) {
  int e = blockIdx.x * blockDim.x + threadIdx.x;
  int cnt = pcnt ? pcnt[0] : fixed_cnt;
  if (e < cnt) {
    int r = rows[e], c = cols[e];
    atomicAdd(&L[(size_t)r * n + c], -dinv[r] * dinv[c]);
  }
}

__global__ void k_L_diag(float* __restrict__ L, int n) {
  int i = blockIdx.x * blockDim.x + threadIdx.x;
  if (i < n) L[(size_t)i * n + i] += 1.0f;
}

__global__ void k_qinit(float* __restrict__ Q, int n) {
  int i = blockIdx.x * blockDim.x + threadIdx.x;
  if (i < n * NEV) {
    unsigned s = (unsigned)i * 2654435761u + 12345u;
    s ^= s >> 13; s *= 2246822519u; s ^= s >> 11;
    Q[i] = ((s & 0xFFFFFF) / 8388608.0f) - 1.0f;
  }
}

// Y[n,16] = L[n,n] * Q[n,16].  One wave per 16-row tile; Q k-panels staged in LDS
// (TDM when available); V_WMMA_F32_16X16X4_F32 accumulation in fp32.
__global__ void __launch_bounds__(256) k_gemm_wmma(const float* __restrict__ L,
                                                   const float* __restrict__ Q,
                                                   float* __restrict__ Y, int n) {
  extern __shared__ float qs[];                 // KP*NEV floats
  const int wave = threadIdx.x >> 5;
  const int lane = threadIdx.x & 31;
  const int m0   = (blockIdx.x * 8 + wave) * 16;
  const int m    = lane & 15;                   // A row within tile
  const int kh   = (lane >> 4) << 1;            // A: lanes 0-15 -> K0/K1, 16-31 -> K2/K3
  const int colc = lane & 15;                   // B/C column
  v8f acc = {};
  for (int k0 = 0; k0 < n; k0 += KP) {
#ifdef HAVE_TDM
    if (threadIdx.x < 32) tdm_load_1d(qs, Q + (size_t)k0 * NEV, KP * NEV);
#else
    for (int idx = threadIdx.x; idx < KP * NEV / 4; idx += 256)
      ((float4*)qs)[idx] = ((const float4*)(Q + (size_t)k0 * NEV))[idx];
#endif
    __syncthreads();
    const float* Lrow = L + (size_t)(m0 + m) * n + k0;
#pragma unroll 4
    for (int kk = 0; kk < KP; kk += 4) {
      if ((kk & 511) == 0) __builtin_prefetch(Lrow + kk + 512, 0, 0);
      v2f a; a.x = Lrow[kk + kh];                 a.y = Lrow[kk + kh + 1];
      v2f b; b.x = qs[(kk + kh) * NEV + colc];    b.y = qs[(kk + kh + 1) * NEV + colc];
      acc = __builtin_amdgcn_wmma_f32_16x16x4_f32(false, a, false, b,
                                                  (short)0, acc, false, false);
    }
    __syncthreads();
  }
  const int mrow = m0 + ((lane >> 4) ? 8 : 0);    // C layout: vgpr i -> M=i (+8 upper lanes)
#pragma unroll
  for (int i = 0; i < 8; i++)
    Y[(size_t)(mrow + i) * NEV + colc] = acc[i];
}

__device__ __forceinline__ float block_reduce(float v, float* red) {
  int t = threadIdx.x;
  red[t] = v; __syncthreads();
  for (int off = 512; off; off >>= 1) { if (t < off) red[t] += red[t + off]; __syncthreads(); }
  float r = red[0]; __syncthreads();
  return r;
}

// Z = use_y ? 2Q - Y : Q; deflate against D^{1/2}1; modified Gram-Schmidt; Q = Z.
// Single workgroup, whole n x 16 block resident in LDS (<= 256KB of the 320KB WGP LDS).
__global__ void __launch_bounds__(1024) k_ortho(float* __restrict__ Q,
                                                const float* __restrict__ Y,
                                                const float* __restrict__ deg,
                                                int n, int use_y) {
  extern __shared__ float sm[];
  float* z   = sm;                        // [n][NEV]
  float* red = sm + (size_t)n * NEV;      // [1024]
  int t = threadIdx.x;
  for (int i = t; i < n * NEV; i += 1024) z[i] = use_y ? (2.0f * Q[i] - Y[i]) : Q[i];
  __syncthreads();
  float part = 0.f;
  for (int i = t; i < n; i += 1024) part += deg[i];
  float wn2 = block_reduce(part, red) + 1e-20f;
  for (int c = 0; c < NEV; c++) {
    part = 0.f;
    for (int i = t; i < n; i += 1024) part += sqrtf(deg[i]) * z[i * NEV + c];
    float coef = block_reduce(part, red) / wn2;
    for (int i = t; i < n; i += 1024) z[i * NEV + c] -= coef * sqrtf(deg[i]);
    __syncthreads();
    for (int p = 0; p < c; p++) {
      part = 0.f;
      for (int i = t; i < n; i += 1024) part += z[i * NEV + p] * z[i * NEV + c];
      float d = block_reduce(part, red);
      for (int i = t; i < n; i += 1024) z[i * NEV + c] -= d * z[i * NEV + p];
      __syncthreads();
    }
    part = 0.f;
    for (int i = t; i < n; i += 1024) { float v = z[i * NEV + c]; part += v * v; }
    float inv = rsqrtf(block_reduce(part, red) + 1e-20f);
    for (int i = t; i < n; i += 1024) z[i * NEV + c] *= inv;
    __syncthreads();
  }
  for (int i = t; i < n * NEV; i += 1024) Q[i] = z[i];
}

__global__ void k_T(const float* __restrict__ Q, const float* __restrict__ Y,
                    float* __restrict__ T, int n) {
  int t = threadIdx.x;            // 256 threads: (p,q)
  int p = t >> 4, q = t & 15;
  float s = 0.f;
  for (int i = 0; i < n; i++) s += Q[(size_t)i * NEV + p] * Y[(size_t)i * NEV + q];
  T[t] = s;
}

__global__ void k_eig_small(const float* __restrict__ T, float* __restrict__ out) {
  if (threadIdx.x != 0) return;
  float a[NEV][NEV];
  for (int p = 0; p < NEV; p++)
    for (int q = 0; q < NEV; q++)
      a[p][q] = 0.5f * (T[p * NEV + q] + T[q * NEV + p]);
  for (int sweep = 0; sweep < 20; sweep++)
    for (int p = 0; p < NEV - 1; p++)
      for (int q = p + 1; q < NEV; q++) {
        float apq = a[p][q];
        if (fabsf(apq) < 1e-12f) continue;
        float theta = 0.5f * (a[q][q] - a[p][p]) / apq;
        float tt = (theta >= 0.f ? 1.f : -1.f) / (fabsf(theta) + sqrtf(theta * theta + 1.f));
        float cc = rsqrtf(tt * tt + 1.f);
        float ss = tt * cc;
        for (int k = 0; k < NEV; k++) {
          float akp = a[k][p], akq = a[k][q];
          a[k][p] = cc * akp - ss * akq; a[k][q] = ss * akp + cc * akq;
        }
        for (int k = 0; k < NEV; k++) {
          float apk = a[p][k], aqk = a[q][k];
          a[p][k] = cc * apk - ss * aqk; a[q][k] = ss * apk + cc * aqk;
        }
      }
  float mn = a[0][0];
  for (int k = 1; k < NEV; k++) mn = fminf(mn, a[k][k]);
  *out = mn;   // smallest deflated Ritz value ~ Fiedler value
}

__global__ void k_tail(const int* __restrict__ perm, const int* __restrict__ offs,
                       const float* __restrict__ fb, const float* __restrict__ fa,
                       float* __restrict__ dout) {
  int j = blockIdx.x * blockDim.x + threadIdx.x;
  int Ecnt = offs[NE];
  size_t base = (size_t)NKEPT * DF + 2 * (size_t)Ecnt;
  if (j < NKEPT) dout[base + j] = (float)(perm[j] / NPERG);
  else if (j == NKEPT) dout[base + NKEPT] = ALPHA_ * fabsf(fb[0] - fa[0]);
}

// ---------------- host orchestration ----------------

extern "C" void kernel_launch(void* const* d_in, const int* in_sizes, int n_in,
                              void* d_out, int out_size, void* d_ws, size_t ws_size,
                              hipStream_t stream) {
  const float* x   = (const float*)d_in[0];
  const int*   ei  = (const int*)d_in[1];
  const int*   row = ei;
  const int*   col = ei + NE;
  const float* W   = (const float*)d_in[3];
  const float* b   = (const float*)d_in[4];
  float* out = (float*)d_out;
  (void)in_sizes; (void)n_in; (void)out_size; (void)ws_size;

  char* w = (char*)d_ws;
  auto carve = [&](size_t bytes) { void* p = (void*)w; w += (bytes + 255) & ~(size_t)255; return p; };
  float* L     = (float*)carve((size_t)NTOT * NTOT * 4);
  float* Q     = (float*)carve((size_t)NTOT * NEV * 4);
  float* Yb    = (float*)carve((size_t)NTOT * NEV * 4);
  float* h     = (float*)carve(NTOT * 4);
  float* degg  = (float*)carve(NTOT * 4);
  float* dinvg = (float*)carve(NTOT * 4);
  float* accs  = (float*)carve(NTOT * 4);
  float* score = (float*)carve(NTOT * 4);
  float* degF  = (float*)carve(NTOT * 4);
  float* dinvF = (float*)carve(NTOT * 4);
  int*   perm  = (int*)carve(NKEPT * 4);
  int*   nid   = (int*)carve(NTOT * 4);
  int*   flags = (int*)carve(NE * 4);
  int*   offs  = (int*)carve((NE + 1) * 4);
  int*   er    = (int*)carve(NE * 4);
  int*   ec    = (int*)carve(NE * 4);
  float* T     = (float*)carve(256 * 4);
  float* fb    = (float*)carve(256);
  float* fa    = (float*)carve(256);

  // ---- GCN score ----
  hipMemsetAsync(degg, 0, NTOT * 4, stream);
  hipMemsetAsync(accs, 0, NTOT * 4, stream);
  k_deg_gcn<<<NE / 256, 256, 0, stream>>>(col, degg);
  k_dinv_gcn<<<NTOT / 256, 256, 0, stream>>>(degg, dinvg);
  k_h<<<NTOT * 32 / 256, 256, 0, stream>>>(x, W, h);
  k_gcn_scatter<<<NE / 256, 256, 0, stream>>>(row, col, dinvg, h, accs);
  k_score<<<NTOT / 256, 256, 0, stream>>>(accs, dinvg, h, b, score);

  // ---- top-k, gather, relabel, edge filter ----
  k_topk<<<NB, 512, 0, stream>>>(score, perm);
  k_gather<<<NKEPT, DF, 0, stream>>>(x, score, perm, out);
  k_newid_init<<<NTOT / 256, 256, 0, stream>>>(nid);
  k_newid_set<<<NKEPT / 256, 256, 0, stream>>>(perm, nid);
  k_flags<<<NE / 256, 256, 0, stream>>>(row, col, nid, flags);
  k_scan<<<1, 1024, 0, stream>>>(flags, offs);
  k_scatter_edges<<<NE / 256, 256, 0, stream>>>(row, col, nid, flags, offs, er, ec,
                                                out + (size_t)NKEPT * DF);

  // ---- Fiedler via deflated block power iteration (WMMA GEMM core) ----
  auto run_fiedler = [&](const int* rows, const int* cols, const int* pcnt, int fixed_cnt,
                         int n, float* fout) {
    hipMemsetAsync(degF, 0, (size_t)n * 4, stream);
    k_deg_edges<<<NE / 256, 256, 0, stream>>>(rows, pcnt, fixed_cnt, degF);
    k_dinv0<<<(n + 255) / 256, 256, 0, stream>>>(degF, dinvF, n);
    hipMemsetAsync(L, 0, (size_t)n * n * 4, stream);
    k_L_edges<<<NE / 256, 256, 0, stream>>>(rows, cols, pcnt, fixed_cnt, dinvF, L, n);
    k_L_diag<<<(n + 255) / 256, 256, 0, stream>>>(L, n);
    k_qinit<<<(n * NEV + 255) / 256, 256, 0, stream>>>(Q, n);
    size_t sh_g = (size_t)KP * NEV * 4;
    size_t sh_o = (size_t)n * NEV * 4 + 1024 * 4;
    k_ortho<<<1, 1024, sh_o, stream>>>(Q, Yb, degF, n, 0);
    for (int it = 0; it < NITER; ++it) {
      k_gemm_wmma<<<n / 128, 256, sh_g, stream>>>(L, Q, Yb, n);
      k_ortho<<<1, 1024, sh_o, stream>>>(Q, Yb, degF, n, 1);
    }
    k_gemm_wmma<<<n / 128, 256, sh_g, stream>>>(L, Q, Yb, n);
    k_T<<<1, 256, 0, stream>>>(Q, Yb, T, n);
    k_eig_small<<<1, 32, 0, stream>>>(T, fout);
  };

  run_fiedler(row, col, nullptr, NE, NTOT, fb);    // before pooling
  run_fiedler(er, ec, offs + NE, 0, NKEPT, fa);    // after pooling (dynamic edge count)

  k_tail<<<(NKEPT + 256) / 256, 256, 0, stream>>>(perm, offs, fb, fa, out);
}